// EfficientMemorySoftmax_39333310497337
// MI455X (gfx1250) — compile-verified
//
#include <hip/hip_runtime.h>
#include <cstdint>
#include <cstddef>

// ---------------------------------------------------------------------------
// Fused softmax (dim=-1) + 64x64 blockwise int8 quantization for
// x: (2,16,2048,2048) fp32  ->  out (fp32 softmax), q (quantized, stored as
// float per harness's single-dtype concat), s, z.
//
// d_out layout (floats): [out: 134217728][q: 134217728][s: 32768][z: 32768]
//
// MI455X reasoning: memory-bound (~1.5 GiB min traffic -> ~65us at 23.3 TB/s).
// No matmul structure => WMMA not applicable; CDNA5-specific path used:
// global_load_async_to_lds_b128 DMA (ASYNCcnt) + s_wait_asynccnt, wave32
// shuffle reductions, NT stores (keep L2 for the half0 re-read), 256KB of the
// WGP's 320KB LDS holding a 64x1024 tile-half in place.
//
// v2: single rolled loop over the two column-halves (unroll 1) to halve the
// B-phase code footprint and kill the scratch spills seen in v1's asm.
// ---------------------------------------------------------------------------

typedef float v4f __attribute__((ext_vector_type(4)));

#define QB       64
#define ROWS     2048
#define COLS     2048
#define RPB      64            // rows per block == QB
#define HALF     1024          // columns per LDS-resident half
#define SROW_F   1028          // padded LDS row stride (floats)
#define SROW_B   (SROW_F * 4)  // 4112 B, 16B aligned
#define OUT_ELEMS 134217728ull // 32 * 2048 * 2048
#define NTILES    32768ull     // 32 * 32 * 32

// Low 32 bits of a generic pointer to LDS == DS-usable byte offset.
__device__ __forceinline__ uint32_t lds_lo32(const void* p) {
  return (uint32_t)(uintptr_t)p;
}

// CDNA5 async DMA: copies 16B per lane from global address sbase + goff to the LDS address in lds.
// Tracked by ASYNCcnt. (CDNA5 ISA, GVS mode)
__device__ __forceinline__ void async_ld_b128(uint32_t lds, uint32_t goff,
                                              const void* sbase) {
  asm volatile("global_load_async_to_lds_b128 %0, %1, %2"
               : : "v"(lds), "v"(goff), "s"(sbase) : "memory");
}

__device__ __forceinline__ void wait_async_zero() {
  asm volatile("s_wait_asynccnt 0x0" : : : "memory");
}

// Online (max, sum-of-exp) update over 4 values; ~1 rescale per float4.
__device__ __forceinline__ void online4(float& m, float& s, v4f v) {
  float m4 = fmaxf(fmaxf(v.x, v.y), fmaxf(v.z, v.w));
  if (m4 > m) { s *= __expf(m - m4); m = m4; }
  s += __expf(v.x - m) + __expf(v.y - m) + __expf(v.z - m) + __expf(v.w - m);
}

__global__ void __launch_bounds__(1024, 1)
softmax_blockquant_fused(const float* __restrict__ x, float* __restrict__ out,
                         float* __restrict__ qo, float* __restrict__ so,
                         float* __restrict__ zo) {
  extern __shared__ char smem[];
  float* ybuf  = (float*)smem;                          // 64 x SROW_F floats
  float* redmn = (float*)(smem + (size_t)RPB * SROW_B); // 32
  float* redmx = redmn + 32;                            // 32
  float* szinv = redmx + 32;                            // 16
  float* szz   = szinv + 16;                            // 16

  const int t  = threadIdx.x;
  const int r  = t >> 4;          // row in block: 0..63
  const int l  = t & 15;          // 16 lanes per row
  const int n  = blockIdx.x >> 5; // batch 0..31
  const int rb = blockIdx.x & 31; // row-block 0..31

  const size_t blk_off = ((size_t)n * ROWS + (size_t)rb * RPB) * COLS;
  const float* xb = x + blk_off;
  float* ob = out + blk_off;

  const uint32_t ldsRow = lds_lo32(ybuf) + (uint32_t)(r * SROW_B);
  const uint32_t growB  = (uint32_t)(r * COLS * 4);

  // Async-stage one column half (h) of this row-block into LDS.
  auto stage = [&](int h) {
#pragma unroll
    for (int i = 0; i < 16; ++i) {
      uint32_t c16 = (uint32_t)((l + i * 16) * 16);
      async_ld_b128(ldsRow + c16, growB + (uint32_t)(h * HALF * 4) + c16, xb);
    }
  };

  // ---- kick off async DMA of half1; it flies during the half0 pass --------
  stage(1);

  // ---- phase A: online row max / sum-of-exp -------------------------------
  float m = -__builtin_huge_valf();
  float s = 0.0f;
  const v4f* xrow = (const v4f*)(xb + (size_t)r * COLS);
#pragma unroll
  for (int i = 0; i < 16; ++i) online4(m, s, xrow[l + i * 16]);  // half0, global

  wait_async_zero();
  __syncthreads();

  const v4f* lrow_c = (const v4f*)((const char*)ybuf + (size_t)r * SROW_B);
#pragma unroll
  for (int i = 0; i < 16; ++i) online4(m, s, lrow_c[l + i * 16]); // half1, LDS

  // butterfly combine across the 16 lanes of this row (wave32 shuffles)
#pragma unroll
  for (int mask = 1; mask <= 8; mask <<= 1) {
    float om = __shfl_xor(m, mask, 32);
    float os = __shfl_xor(s, mask, 32);
    float nm = fmaxf(m, om);
    s = s * __expf(m - nm) + os * __expf(om - nm);
    m = nm;
  }
  const float inv = 1.0f / s;

  // ---- phase B: one rolled pass per half (h=1 in LDS now, then h=0) -------
#pragma unroll 1
  for (int h = 1; h >= 0; --h) {
    // B1: y = exp(x-m)*inv ; in-place in LDS + NT store to out
    v4f* lrw  = (v4f*)((char*)ybuf + (size_t)r * SROW_B);
    v4f* orow = (v4f*)(ob + (size_t)r * COLS) + h * (HALF / 4);
#pragma unroll
    for (int i = 0; i < 16; ++i) {
      int f4 = l + i * 16;
      v4f v = lrw[f4];
      v4f y;
      y.x = __expf(v.x - m) * inv;
      y.y = __expf(v.y - m) * inv;
      y.z = __expf(v.z - m) * inv;
      y.w = __expf(v.w - m) * inv;
      lrw[f4] = y;
      __builtin_nontemporal_store(y, orow + f4);
    }
    __syncthreads();

    // B2: tile min/max -- 64 threads per 64x64 tile, each 4 rows x 16 cols
    const int g = t >> 6, k = t & 63, rg = k >> 2, scn = k & 3;
    float mn = __builtin_huge_valf(), mx = -__builtin_huge_valf();
#pragma unroll
    for (int ii = 0; ii < 4; ++ii) {
      const v4f* p = (const v4f*)((const char*)ybuf +
                                  (size_t)(rg * 4 + ii) * SROW_B) +
                     g * 16 + scn * 4;
#pragma unroll
      for (int jj = 0; jj < 4; ++jj) {
        v4f v = p[jj];
        mn = fminf(mn, fminf(fminf(v.x, v.y), fminf(v.z, v.w)));
        mx = fmaxf(mx, fmaxf(fmaxf(v.x, v.y), fmaxf(v.z, v.w)));
      }
    }
#pragma unroll
    for (int mask = 1; mask <= 16; mask <<= 1) {
      mn = fminf(mn, __shfl_xor(mn, mask, 32));
      mx = fmaxf(mx, __shfl_xor(mx, mask, 32));
    }
    if ((t & 31) == 0) { redmn[t >> 5] = mn; redmx[t >> 5] = mx; }
    __syncthreads();
    if (k == 0) {  // one thread per tile: combine the tile's 2 waves, emit s,z
      float tmn = fminf(redmn[2 * g], redmn[2 * g + 1]);
      float tmx = fmaxf(redmx[2 * g], redmx[2 * g + 1]);
      float sc  = (tmx - tmn) * (1.0f / 255.0f);
      float d   = sc + 1e-10f;
      float invd = 1.0f / d;
      float z   = -tmn * invd - 128.0f;
      int blk   = (n * 32 + rb) * 32 + h * 16 + g;
      so[blk] = sc;
      zo[blk] = z;
      szinv[g] = invd;
      szz[g]   = z;
    }
    __syncthreads();

    // B3: quantize with the linear mapping -> fully coalesced q stores
#pragma unroll
    for (int i = 0; i < 16; ++i) {
      float invd = szinv[i], z = szz[i];
      v4f v = lrw[l + i * 16];
      v4f q;
      q.x = rintf(fminf(fmaxf(v.x * invd + z, -128.0f), 127.0f));
      q.y = rintf(fminf(fmaxf(v.y * invd + z, -128.0f), 127.0f));
      q.z = rintf(fminf(fmaxf(v.z * invd + z, -128.0f), 127.0f));
      q.w = rintf(fminf(fmaxf(v.w * invd + z, -128.0f), 127.0f));
      size_t blk = (size_t)((n * 32 + rb) * 32 + h * 16 + i);
      __builtin_nontemporal_store(
          q, (v4f*)(qo + blk * (size_t)(QB * QB) + r * QB + l * 4));
    }
    __syncthreads();  // ybuf free for reuse

    // Re-stage half0 (should be L2-resident: NT stores above kept L2 clean)
    if (h == 1) {
      stage(0);
      wait_async_zero();
      __syncthreads();
    }
  }
}

extern "C" void kernel_launch(void* const* d_in, const int* in_sizes, int n_in,
                              void* d_out, int out_size, void* d_ws, size_t ws_size,
                              hipStream_t stream) {
  (void)in_sizes; (void)n_in; (void)out_size; (void)d_ws; (void)ws_size;

  const float* x = (const float*)d_in[0];
  float* out = (float*)d_out;
  float* qo  = out + OUT_ELEMS;
  float* so  = qo + OUT_ELEMS;
  float* zo  = so + NTILES;

  const size_t shmem = (size_t)RPB * SROW_B + 96 * sizeof(float); // 263552 B

  // Allow >default dynamic LDS (WGP has 320KB). Deterministic, idempotent.
  (void)hipFuncSetAttribute((const void*)softmax_blockquant_fused,
                            hipFuncAttributeMaxDynamicSharedMemorySize,
                            (int)shmem);

  softmax_blockquant_fused<<<dim3(32 * 32), dim3(1024), shmem, stream>>>(
      x, out, qo, so, zo);
}